// GCNEncoder_43971875177077
// MI455X (gfx1250) — compile-verified
//
#include <hip/hip_runtime.h>
#include <hip/hip_bf16.h>

// GCN 3-layer encoder for MI455X (gfx1250, wave32).
// GEMMs use V_WMMA_F32_16X16X4_F32 (full f32 precision, matches reference).
// Edge aggregation: one wave32 per edge, coalesced row gather + HW f32 atomics.

typedef float v2f __attribute__((ext_vector_type(2)));
typedef float v8f __attribute__((ext_vector_type(8)));

static constexpr int IN_C = 128;
static constexpr int HIDF = 128;
static constexpr int OUTF = 64;
static constexpr float EPSV = 1e-5f;

// ---------------- degree / dinv ----------------

__global__ void zero_f32_kernel(float* __restrict__ p, int n) {
    int i = blockIdx.x * blockDim.x + threadIdx.x;
    if (i < n) p[i] = 0.0f;
}

__global__ void deg_count_kernel(const long long* __restrict__ dst,
                                 float* __restrict__ deg, int E) {
    int i = blockIdx.x * blockDim.x + threadIdx.x;
    if (i < E) unsafeAtomicAdd(&deg[(int)dst[i]], 1.0f);
}

__global__ void deg_to_dinv_kernel(float* __restrict__ d, int n) {
    int i = blockIdx.x * blockDim.x + threadIdx.x;
    if (i < n) d[i] = rsqrtf(1.0f + d[i]);
}

// ---------------- GEMM: H[nrows x FOUT] = A[nrows x 128] @ W[128 x FOUT] ---------------
// One wave per 16x16 output tile; K loop of 32 x V_WMMA_F32_16X16X4_F32.

template <int FOUT>
__global__ void gemm_wmma_f32_kernel(const float* __restrict__ A,
                                     const float* __restrict__ W,
                                     float* __restrict__ H, int nrows) {
    constexpr int K = 128;
    constexpr int CT = FOUT / 16;
    int wave = (int)((blockIdx.x * blockDim.x + threadIdx.x) >> 5);
    int lane = (int)(threadIdx.x & 31);
    int rowTile = wave / CT;
    int colTile = wave % CT;
    int rowBase = rowTile * 16;
    if (rowBase >= nrows) return;           // wave-uniform: EXEC stays all-ones
    int colBase = colTile * 16;

    int m    = lane & 15;                   // A row / B,C column within tile
    int half = lane >> 4;                   // lane half selects K pair / M half
    int kh   = half * 2;

    int arow = rowBase + m;
    if (arow >= nrows) arow = nrows - 1;    // clamp (safe read; stores guarded below)
    const float* Arow = A + (size_t)arow * K;
    const float* Wp   = W + colBase + m;    // column (N) = m

    v8f acc = {};
#pragma unroll
    for (int k0 = 0; k0 < K; k0 += 4) {
        v2f a, b;
        a.x = Arow[k0 + kh];
        a.y = Arow[k0 + kh + 1];
        b.x = Wp[(size_t)(k0 + kh)     * FOUT];
        b.y = Wp[(size_t)(k0 + kh + 1) * FOUT];
        acc = __builtin_amdgcn_wmma_f32_16x16x4_f32(
            /*neg_a=*/false, a, /*neg_b=*/false, b,
            /*c_mod=*/(short)0, acc, /*reuse_a=*/false, /*reuse_b=*/false);
    }

    float* Hp = H + colBase + m;
    int mBase = rowBase + half * 8;
#pragma unroll
    for (int v = 0; v < 8; ++v) {
        int row = mBase + v;
        if (row < nrows) Hp[(size_t)row * FOUT] = acc[v];
    }
}

// ---------------- self-loop init: agg = h * dinv^2 ----------------

__global__ void self_loop_init_kernel(const float* __restrict__ h,
                                      const float* __restrict__ dinv,
                                      float* __restrict__ agg, int n, int F) {
    int i = blockIdx.x * blockDim.x + threadIdx.x;
    if (i < n * F) {
        float di = dinv[i / F];
        agg[i] = h[i] * di * di;
    }
}

// ---------------- edge scatter: agg[dst] += h[src] * dinv[src]*dinv[dst] ---------------
// One wave32 per edge; each lane owns F/32 consecutive floats of the row
// (row load is a fully coalesced 512B / 256B burst across the wave).

template <int F>
__global__ void edge_scatter_kernel(const long long* __restrict__ src,
                                    const long long* __restrict__ dst,
                                    const float* __restrict__ h,
                                    const float* __restrict__ dinv,
                                    float* __restrict__ agg, int E) {
    int wave = (int)((blockIdx.x * blockDim.x + threadIdx.x) >> 5);
    int lane = (int)(threadIdx.x & 31);
    if (wave >= E) return;
    int s = (int)src[wave];
    int d = (int)dst[wave];
    float c = dinv[s] * dinv[d];
    constexpr int EPL = F / 32;             // 4 (F=128) or 2 (F=64)
    const float* hp = h + (size_t)s * F + lane * EPL;
    float* ap = agg + (size_t)d * F + lane * EPL;
    if constexpr (EPL == 4) {
        float4 v = *(const float4*)hp;      // global_load_b128
        unsafeAtomicAdd(ap + 0, v.x * c);
        unsafeAtomicAdd(ap + 1, v.y * c);
        unsafeAtomicAdd(ap + 2, v.z * c);
        unsafeAtomicAdd(ap + 3, v.w * c);
    } else {
        float2 v = *(const float2*)hp;      // global_load_b64
        unsafeAtomicAdd(ap + 0, v.x * c);
        unsafeAtomicAdd(ap + 1, v.y * c);
    }
}

// ---------------- bias + BN(eval) + leaky(0.1) ----------------

__global__ void bias_bn_leaky_kernel(const float* __restrict__ agg,
                                     const float* __restrict__ b,
                                     const float* __restrict__ g,
                                     const float* __restrict__ be,
                                     const float* __restrict__ mean,
                                     const float* __restrict__ var,
                                     float* __restrict__ out, int n, int F) {
    int i = blockIdx.x * blockDim.x + threadIdx.x;
    if (i < n * F) {
        int c = i & (F - 1);                // F is a power of two
        float scale = g[c] * rsqrtf(var[c] + EPSV);
        float x = agg[i] + b[c];
        float y = (x - mean[c]) * scale + be[c];
        out[i] = (y > 0.0f) ? y : 0.1f * y;
    }
}

__global__ void add_bias_kernel(const float* __restrict__ agg,
                                const float* __restrict__ b,
                                float* __restrict__ out, int n, int F) {
    int i = blockIdx.x * blockDim.x + threadIdx.x;
    if (i < n * F) out[i] = agg[i] + b[i & (F - 1)];
}

// ---------------- launch ----------------

static inline int cdiv_i(int a, int b) { return (a + b - 1) / b; }

extern "C" void kernel_launch(void* const* d_in, const int* in_sizes, int n_in,
                              void* d_out, int out_size, void* d_ws, size_t ws_size,
                              hipStream_t stream) {
    const float*     x   = (const float*)d_in[0];
    const long long* ei  = (const long long*)d_in[1];   // int64 [2, E]
    const float*     W1  = (const float*)d_in[2];
    const float*     b1  = (const float*)d_in[3];
    const float*     g1  = (const float*)d_in[4];
    const float*     be1 = (const float*)d_in[5];
    const float*     m1  = (const float*)d_in[6];
    const float*     v1  = (const float*)d_in[7];
    const float*     W2  = (const float*)d_in[8];
    const float*     b2  = (const float*)d_in[9];
    const float*     g2  = (const float*)d_in[10];
    const float*     be2 = (const float*)d_in[11];
    const float*     m2  = (const float*)d_in[12];
    const float*     v2  = (const float*)d_in[13];
    const float*     W3  = (const float*)d_in[14];
    const float*     b3  = (const float*)d_in[15];
    float*           out = (float*)d_out;

    const int N = in_sizes[0] / IN_C;
    const int E = in_sizes[1] / 2;
    const long long* src = ei;
    const long long* dst = ei + E;

    float* ws   = (float*)d_ws;
    float* dinv = ws;                              // N
    float* h    = dinv + N;                        // N*128
    float* agg  = h + (size_t)N * HIDF;            // N*128
    float* act  = agg + (size_t)N * HIDF;          // N*128

    const dim3 blk(256);

    // degrees (graph-invariant: compute once, reuse for all 3 layers)
    zero_f32_kernel<<<cdiv_i(N, 256), blk, 0, stream>>>(dinv, N);
    deg_count_kernel<<<cdiv_i(E, 256), blk, 0, stream>>>(dst, dinv, E);
    deg_to_dinv_kernel<<<cdiv_i(N, 256), blk, 0, stream>>>(dinv, N);

    const int waves128 = cdiv_i(N, 16) * (HIDF / 16);
    const int waves64  = cdiv_i(N, 16) * (OUTF / 16);
    const int edgeBlks = cdiv_i(E * 32, 256);

    // ---- layer 1 ----
    gemm_wmma_f32_kernel<HIDF><<<cdiv_i(waves128 * 32, 256), blk, 0, stream>>>(x, W1, h, N);
    self_loop_init_kernel<<<cdiv_i(N * HIDF, 256), blk, 0, stream>>>(h, dinv, agg, N, HIDF);
    edge_scatter_kernel<HIDF><<<edgeBlks, blk, 0, stream>>>(src, dst, h, dinv, agg, E);
    bias_bn_leaky_kernel<<<cdiv_i(N * HIDF, 256), blk, 0, stream>>>(agg, b1, g1, be1, m1, v1, act, N, HIDF);

    // ---- layer 2 ----
    gemm_wmma_f32_kernel<HIDF><<<cdiv_i(waves128 * 32, 256), blk, 0, stream>>>(act, W2, h, N);
    self_loop_init_kernel<<<cdiv_i(N * HIDF, 256), blk, 0, stream>>>(h, dinv, agg, N, HIDF);
    edge_scatter_kernel<HIDF><<<edgeBlks, blk, 0, stream>>>(src, dst, h, dinv, agg, E);
    bias_bn_leaky_kernel<<<cdiv_i(N * HIDF, 256), blk, 0, stream>>>(agg, b2, g2, be2, m2, v2, act, N, HIDF);

    // ---- layer 3 (no BN / activation, just bias) ----
    gemm_wmma_f32_kernel<OUTF><<<cdiv_i(waves64 * 32, 256), blk, 0, stream>>>(act, W3, h, N);
    self_loop_init_kernel<<<cdiv_i(N * OUTF, 256), blk, 0, stream>>>(h, dinv, agg, N, OUTF);
    edge_scatter_kernel<OUTF><<<edgeBlks, blk, 0, stream>>>(src, dst, h, dinv, agg, E);
    add_bias_kernel<<<cdiv_i(N * OUTF, 256), blk, 0, stream>>>(agg, b3, out, N, OUTF);
}